// ABLE_48198122995803
// MI455X (gfx1250) — compile-verified
//
#include <hip/hip_runtime.h>
#include <math.h>

#define B     2048
#define C     1000
#define D     256
#define TWOB  4096
#define TEMP       0.1f
#define BASE_TEMP  0.07f

typedef __attribute__((ext_vector_type(2))) float v2f;
typedef __attribute__((ext_vector_type(8))) float v8f;

// cf = concat(features[:,0,:], features[:,1,:]) ; features is (B,2,D) row-major.
__device__ __forceinline__ const float* cf_row(const float* feats, int r) {
    return feats + (size_t)(((r & (B - 1)) * 2 + (r >> 11)) * D);
}

// ---------------------------------------------------------------------------
// Kernel 1: per-row masked argmax (tp), and new_target = Y*exp(o-m)/sum_Y exp(o-m)
// ---------------------------------------------------------------------------
__global__ void k_row_softmax(const float* __restrict__ outputs,
                              const float* __restrict__ Y,
                              float* __restrict__ new_target,
                              int* __restrict__ tp) {
    __shared__ float smax[256];
    __shared__ int   sidx[256];
    __shared__ float ssum[256];
    const int row = blockIdx.x;
    const float* o = outputs + (size_t)row * C;
    const float* y = Y + (size_t)row * C;
    const int t = threadIdx.x;

    float mv = -__builtin_inff();
    int   mi = C;
    for (int c = t; c < C; c += 256) {
        if (y[c] != 0.0f) {
            float v = o[c];
            if (v > mv || (v == mv && c < mi)) { mv = v; mi = c; }
        }
    }
    smax[t] = mv; sidx[t] = mi;
    __syncthreads();
    for (int s = 128; s > 0; s >>= 1) {
        if (t < s) {
            float ov = smax[t + s]; int oi = sidx[t + s];
            if (ov > smax[t] || (ov == smax[t] && oi < sidx[t])) { smax[t] = ov; sidx[t] = oi; }
        }
        __syncthreads();
    }
    const float m = smax[0];
    const int best = sidx[0];

    float ls = 0.0f;
    for (int c = t; c < C; c += 256)
        if (y[c] != 0.0f) ls += expf(o[c] - m);
    ssum[t] = ls;
    __syncthreads();
    for (int s = 128; s > 0; s >>= 1) { if (t < s) ssum[t] += ssum[t + s]; __syncthreads(); }
    const float inv = 1.0f / ssum[0];

    float* nt = new_target + (size_t)row * C;
    for (int c = t; c < C; c += 256)
        nt[c] = (y[c] != 0.0f) ? expf(o[c] - m) * inv : 0.0f;
    if (t == 0) tp[row] = best;
}

// ---------------------------------------------------------------------------
// counts
// ---------------------------------------------------------------------------
__global__ void k_zero_counts(int* __restrict__ counts) {
    int t = blockIdx.x * blockDim.x + threadIdx.x;
    if (t < C) counts[t] = 0;
}

__global__ void k_counts(const int* __restrict__ tp, int* __restrict__ counts) {
    int i = blockIdx.x * blockDim.x + threadIdx.x;
    if (i < B) atomicAdd(&counts[tp[i]], 1);
}

// ---------------------------------------------------------------------------
// Wg[i*B + b] = W[i, tp[b]] = Y[i,g]*ps[i,g]/counts[g]   (counts[g] > 0 by construction)
// ---------------------------------------------------------------------------
__global__ void k_build_wg(const float* __restrict__ Y,
                           const float* __restrict__ ps,
                           const int* __restrict__ tp,
                           const int* __restrict__ counts,
                           float* __restrict__ Wg) {
    long long idx = (long long)blockIdx.x * blockDim.x + threadIdx.x;
    if (idx >= (long long)B * B) return;
    int i = (int)(idx >> 11);        // / B
    int b = (int)(idx & (B - 1));    // % B
    int g = tp[b];
    float cnt = (float)counts[g];
    Wg[idx] = Y[(size_t)i * C + g] * ps[(size_t)i * C + g] / cnt;
}

// ---------------------------------------------------------------------------
// Fused WMMA Gram + online logsumexp + weighted contraction.
// One wave per 16-row block of cf (256 waves total). A-fragments for the full
// K=256 held in registers; B tiles streamed from L2. Per 16x16 tile: 64
// chained v_wmma_f32_16x16x4_f32, then per-element online stats.
// ---------------------------------------------------------------------------
__global__ void __launch_bounds__(256) k_gram(const float* __restrict__ feats,
                                              const float* __restrict__ Wg,
                                              float* __restrict__ mlpp) {
    const int lane = threadIdx.x & 31;
    const int wave = (blockIdx.x * blockDim.x + threadIdx.x) >> 5;
    const int m = lane & 15;   // N index within tile (and M index for A loads)
    const int h = lane >> 4;   // lane-half
    const int i0 = wave * 16;

    // A fragments: lane (m,h) holds A[m][k+2h], A[m][k+2h+1] for each K-chunk k.
    const float* arow = cf_row(feats, i0 + m);
    v2f af[64];
#pragma unroll
    for (int kk = 0; kk < 64; ++kk)
        af[kk] = *(const v2f*)(arow + 4 * kk + 2 * h);

    float runMax[8], runSum[8], t1[8], t2[8];
#pragma unroll
    for (int r = 0; r < 8; ++r) {
        runMax[r] = -__builtin_inff();
        runSum[r] = 0.0f; t1[r] = 0.0f; t2[r] = 0.0f;
    }

    for (int jt = 0; jt < TWOB / 16; ++jt) {
        const int j0 = jt * 16;
        const float* brow = cf_row(feats, j0 + m);

        v8f c = {0.f, 0.f, 0.f, 0.f, 0.f, 0.f, 0.f, 0.f};
#pragma unroll
        for (int kk = 0; kk < 64; ++kk) {
            v2f bf = *(const v2f*)(brow + 4 * kk + 2 * h);
            c = __builtin_amdgcn_wmma_f32_16x16x4_f32(
                    /*neg_a=*/false, af[kk], /*neg_b=*/false, bf,
                    /*c_mod=*/(short)0, c, /*reuse_a=*/false, /*reuse_b=*/false);
        }

        const int gj = j0 + m;              // global column (N = lane%16 in both halves)
        const int wcol = gj & (B - 1);
#pragma unroll
        for (int r = 0; r < 8; ++r) {
            const int gi = i0 + r + 8 * h;  // global row
            float s = c[r] * (1.0f / TEMP); // adc value
            bool diag = (gi == gj);
            float w = diag ? 0.0f : Wg[(size_t)(gi & (B - 1)) * B + wcol];
            t1[r] += w * s;
            t2[r] += w;
            // online max (diagonal included in max, excluded from sumexp)
            if (s > runMax[r]) { runSum[r] *= expf(runMax[r] - s); runMax[r] = s; }
            if (!diag) runSum[r] += expf(s - runMax[r]);
        }
    }

    // Reduce across the 16 lanes of each half (wave32 butterflies, masks 1..8).
#pragma unroll
    for (int r = 0; r < 8; ++r) {
        for (int msk = 1; msk < 16; msk <<= 1) {
            float om  = __shfl_xor(runMax[r], msk, 32);
            float os  = __shfl_xor(runSum[r], msk, 32);
            float o1  = __shfl_xor(t1[r], msk, 32);
            float o2  = __shfl_xor(t2[r], msk, 32);
            float nm  = fmaxf(runMax[r], om);
            runSum[r] = runSum[r] * expf(runMax[r] - nm) + os * expf(om - nm);
            runMax[r] = nm;
            t1[r] += o1;
            t2[r] += o2;
        }
        if (m == 0) {
            float v = (t1[r] - runMax[r] * t2[r]) - logf(runSum[r]) * t2[r];
            mlpp[i0 + r + 8 * h] = v;
        }
    }
}

// ---------------------------------------------------------------------------
// Deterministic final reduce: loss = -(T/BT) * mean(mlpp)
// ---------------------------------------------------------------------------
__global__ void k_loss(const float* __restrict__ mlpp, float* __restrict__ out) {
    __shared__ float sh[1024];
    int t = threadIdx.x;
    float v = mlpp[t] + mlpp[t + 1024] + mlpp[t + 2048] + mlpp[t + 3072];
    sh[t] = v;
    __syncthreads();
    for (int s = 512; s > 0; s >>= 1) { if (t < s) sh[t] += sh[t + s]; __syncthreads(); }
    if (t == 0) out[0] = -(TEMP / BASE_TEMP) * (sh[0] / (float)TWOB);
}

// ---------------------------------------------------------------------------
extern "C" void kernel_launch(void* const* d_in, const int* in_sizes, int n_in,
                              void* d_out, int out_size, void* d_ws, size_t ws_size,
                              hipStream_t stream) {
    const float* outputs  = (const float*)d_in[0];  // (2B, C)
    const float* features = (const float*)d_in[1];  // (B, 2, D)
    const float* Y        = (const float*)d_in[2];  // (B, C)
    const float* ps       = (const float*)d_in[3];  // (B, C)
    float* out = (float*)d_out;                     // [loss, new_target(B*C)]

    char* ws = (char*)d_ws;
    int*   tp     = (int*)(ws);                 //  2048 ints
    int*   counts = (int*)(ws + 8192);          //  1000 ints
    float* mlpp   = (float*)(ws + 12288);       //  4096 floats
    float* Wg     = (float*)(ws + 32768);       //  2048*2048 floats (16 MB)

    k_row_softmax<<<B, 256, 0, stream>>>(outputs, Y, out + 1, tp);
    k_zero_counts<<<(C + 255) / 256, 256, 0, stream>>>(counts);
    k_counts<<<(B + 255) / 256, 256, 0, stream>>>(tp, counts);
    k_build_wg<<<(B * B) / 256, 256, 0, stream>>>(Y, ps, tp, counts, Wg);
    k_gram<<<32, 256, 0, stream>>>(features, Wg, mlpp);
    k_loss<<<1, 1024, 0, stream>>>(mlpp, out);
}